// Gaussian2DRasterizer_33560874451062
// MI455X (gfx1250) — compile-verified
//
#include <hip/hip_runtime.h>
#include <math.h>

// ---------------------------------------------------------------------------
// 2D Gaussian splat rasterizer for MI455X (gfx1250, wave32).
//   Kernel 1: single-block bitonic depth argsort (LDS).
//   Kernel 2: one 16x16-pixel tile per 256-thread block; per 256-gaussian
//             chunk: conservative tile cull + order-preserving LDS compaction,
//             then 16-gaussian groups evaluated with V_WMMA_F32_16X16X4_F32
//             (power = rank-6 polynomial matmul in tile-local coords), serial
//             front-to-back alpha blend per lane, block-uniform early-out.
//   Half-wave redistribution of the WMMA D-tile uses v_permlanex16_b32
//   (VALU) instead of ds_bpermute to keep LDS bandwidth for the b128
//   parameter broadcasts.
// ---------------------------------------------------------------------------

#define IMG_W 512
#define IMG_H 512
#define SORTN 8192
#define ALPHA_MIN (1.0f / 255.0f)
#define ALPHA_MAX 0.99f
#define BLUR 0.3f

typedef float v2f __attribute__((ext_vector_type(2)));
typedef float v8f __attribute__((ext_vector_type(8)));

static __device__ __forceinline__ v8f wmma4(v2f a, v2f b, v8f c) {
  // D = A(16x4) * B(4x16) + C ;  f32 everywhere
  return __builtin_amdgcn_wmma_f32_16x16x4_f32(false, a, false, b, (short)0, c,
                                               false, false);
}

// lane L <-> lane L^16 exchange within a wave32
static __device__ __forceinline__ float xswap16(float v) {
#if __has_builtin(__builtin_amdgcn_permlanex16)
  unsigned u = __float_as_uint(v);
  unsigned r = __builtin_amdgcn_permlanex16(u, u, 0x76543210u, 0xfedcba98u,
                                            false, false);
  return __uint_as_float(r);
#else
  return __shfl_xor(v, 16, 32);
#endif
}

// ------------------------- depth argsort (stable-ish) -----------------------
__global__ __launch_bounds__(1024) void gs_sort(const float* __restrict__ depth,
                                                int n, unsigned* __restrict__ order) {
  __shared__ float sk[SORTN];
  __shared__ unsigned short sv[SORTN];
  const int tid = threadIdx.x;
  for (int t = tid; t < SORTN; t += 1024) {
    sk[t] = (t < n) ? depth[t] : 3.0e38f;
    sv[t] = (unsigned short)t;
  }
  __syncthreads();
  for (int k = 2; k <= SORTN; k <<= 1) {
    for (int j = k >> 1; j > 0; j >>= 1) {
      for (int t = tid; t < SORTN; t += 1024) {
        int ixj = t ^ j;
        if (ixj > t) {
          float a = sk[t], b = sk[ixj];
          unsigned short va = sv[t], vb = sv[ixj];
          bool up = ((t & k) == 0);
          bool gt = (a > b) || (a == b && va > vb);   // tie-break: stable
          if (gt == up) { sk[t] = b; sk[ixj] = a; sv[t] = vb; sv[ixj] = va; }
        }
      }
      __syncthreads();
    }
  }
  for (int t = tid; t < n; t += 1024) order[t] = sv[t];
}

// ------------------------------ rasterizer ----------------------------------
__global__ __launch_bounds__(256) void gs_raster(
    const float* __restrict__ means, const float* __restrict__ colors,
    const float* __restrict__ opac, const float* __restrict__ scales,
    const float* __restrict__ rots, const unsigned* __restrict__ order,
    const float* __restrict__ bg, float* __restrict__ out, int n) {
  // chunk staging (SoA), order-compacted survivors
  __shared__ float sc0[256], sc1[256], sc2[256], sc3[256], sc4[256], sc5[256];
  __shared__ float sOp[256], sCr[256], sCg[256], sCb[256];
  __shared__ int swt[8];
  __shared__ float sred[8];

  const int tid = threadIdx.x;
  const int lane = tid & 31;
  const int wv = tid >> 5;
  const bool lo = lane < 16;

  const int tileX = blockIdx.x & 31;         // 512/16 = 32 tiles per row
  const int tileY = blockIdx.x >> 5;
  const float tcx = tileX * 16 + 8.0f;       // tile center (local-coord origin)
  const float tcy = tileY * 16 + 8.0f;

  // this lane's pixel: wave w owns rows 2w, 2w+1; 16 columns per row
  const int col = lane & 15;
  const int row = (wv << 1) + (lane >> 4);

  // B operands (pixel features, loop-invariant):
  //   K0..3 = [x^2, y^2, x*y, x],  K4..7 = [y, 1, 0, 0]   (tile-local coords)
  // layout: VGPR0 = {K0 lanes0-15 | K2 lanes16-31}, VGPR1 = {K1 | K3}
  const float qx = (float)col - 7.5f;
  const float r0 = (float)(wv << 1) - 7.5f;   // pixel-row 0 of this wave
  const float r1 = r0 + 1.0f;                 // pixel-row 1
  v2f B1a, B2a, B1b, B2b;
  B1a.x = lo ? qx * qx : qx * r0;  B1a.y = lo ? r0 * r0 : qx;
  B2a.x = lo ? r0 : 0.0f;          B2a.y = lo ? 1.0f : 0.0f;
  B1b.x = lo ? qx * qx : qx * r1;  B1b.y = lo ? r1 * r1 : qx;
  B2b.x = lo ? r1 : 0.0f;          B2b.y = lo ? 1.0f : 0.0f;

  // tile rectangle of pixel centers (for conservative cull)
  const float rlx = tileX * 16 + 0.5f, rhx = rlx + 15.0f;
  const float rly = tileY * 16 + 0.5f, rhy = rly + 15.0f;

  float T = 1.0f, aR = 0.0f, aG = 0.0f, aB = 0.0f;

  for (int base = 0; base < n; base += 256) {
    if (base + 256 < n)
      __builtin_prefetch(order + base + 256, 0, 0);   // global_prefetch_b8

    // ---- per-thread: derive params for gaussian (base+tid), cull vs tile ---
    const int i = base + tid;
    const int ii = (i < n) ? i : (n - 1);
    const unsigned g = order[ii];
    const float mx = means[2 * g], my = means[2 * g + 1];
    const float sx = scales[2 * g], sy = scales[2 * g + 1];
    const float th = rots[g];
    const float o = opac[g];
    const float cr = colors[3 * g], cg = colors[3 * g + 1], cb = colors[3 * g + 2];

    float sth, cth;
    __sincosf(th, &sth, &cth);
    const float sx2 = sx * sx, sy2 = sy * sy;
    const float Ac = cth * cth * sx2 + sth * sth * sy2;
    const float Cc = sth * sth * sx2 + cth * cth * sy2;
    const float Bc = cth * sth * (sx2 - sy2);
    const float det = Ac * Cc - Bc * Bc;
    const float A2 = Ac + BLUR, C2 = Cc + BLUR;
    const float det2 = A2 * C2 - Bc * Bc;
    const float oadj = o * sqrtf(fmaxf(det / det2, 0.0f));
    const float invd = 1.0f / det2;
    const float ia = C2 * invd, ib = -Bc * invd, ic = A2 * invd;

    // conservative radius: alpha_max <= oadj * exp(-0.5 d^2 / lammax)
    const float lmx = 0.5f * ((A2 + C2) + sqrtf((A2 - C2) * (A2 - C2) + 4.0f * Bc * Bc));
    const float t255 = oadj * 255.0f;
    const float rad2 = (t255 > 1.0f) ? 2.0f * lmx * (__logf(t255) + 1e-4f) : -1.0f;
    const float ddx = fmaxf(0.0f, fmaxf(rlx - mx, mx - rhx));
    const float ddy = fmaxf(0.0f, fmaxf(rly - my, my - rhy));
    const bool flag = (i < n) && (ddx * ddx + ddy * ddy <= rad2);

    // ---- order-preserving compaction (ballot scan across 8 waves) ---------
    const unsigned mask = (unsigned)__ballot(flag ? 1 : 0);
    const int rank = __popc(mask & ((1u << lane) - 1u));
    if (lane == 0) swt[wv] = __popc(mask);
    __syncthreads();
    int off = 0, count = 0;
#pragma unroll
    for (int w2 = 0; w2 < 8; ++w2) {
      const int v = swt[w2];
      if (w2 < wv) off += v;
      count += v;
    }
    if (flag) {
      const int slot = off + rank;
      const float mxl = mx - tcx, myl = my - tcy;   // tile-local mean
      sc0[slot] = -0.5f * ia;
      sc1[slot] = -0.5f * ic;
      sc2[slot] = -ib;
      sc3[slot] = ia * mxl + ib * myl;
      sc4[slot] = ic * myl + ib * mxl;
      sc5[slot] = -0.5f * (ia * mxl * mxl + ic * myl * myl) - ib * mxl * myl;
      sOp[slot] = oadj;
      sCr[slot] = cr; sCg[slot] = cg; sCb[slot] = cb;
    }
    if (tid >= count) {  // zero-pad so partial 16-groups are inert (op = 0)
      sc0[tid] = 0.0f; sc1[tid] = 0.0f; sc2[tid] = 0.0f;
      sc3[tid] = 0.0f; sc4[tid] = 0.0f; sc5[tid] = 0.0f;
      sOp[tid] = 0.0f; sCr[tid] = 0.0f; sCg[tid] = 0.0f; sCb[tid] = 0.0f;
    }
    __syncthreads();

    // ---- process survivors in 16-gaussian groups via WMMA -----------------
    const int ngroups = (count + 15) >> 4;
    for (int gi = 0; gi < ngroups; ++gi) {
      const int g0 = gi << 4;
      const int m = g0 + (lane & 15);
      // A (coefficients, 16x4): VGPR0={K0|K2}, VGPR1={K1|K3}
      v2f A1, A2v;
      A1.x = lo ? sc0[m] : sc2[m];
      A1.y = lo ? sc1[m] : sc3[m];
      A2v.x = lo ? sc4[m] : 0.0f;
      A2v.y = lo ? sc5[m] : 0.0f;

      v8f z = {};
      v8f D1 = wmma4(A2v, B2a, z);
      D1 = wmma4(A1, B1a, D1);     // powers: 16 gaussians x 16 pixels (row 2w)
      v8f D2 = wmma4(A2v, B2b, z);
      D2 = wmma4(A1, B1b, D2);     // powers for row 2w+1

      // redistribute: each lane gathers all 16 powers of its own pixel.
      // lane L<16 owns D1 col L (has rows 0-7, needs 8-15 from lane L+16);
      // lane L>=16 owns D2 col L-16 (has rows 8-15, needs 0-7 from lane L-16).
      float pw[16];
#pragma unroll
      for (int t2 = 0; t2 < 8; ++t2) {
        const float send = lo ? D2[t2] : D1[t2];
        const float recv = xswap16(send);            // v_permlanex16_b32
        const float own = lo ? D1[t2] : D2[t2];
        pw[t2] = lo ? own : recv;
        pw[8 + t2] = lo ? recv : own;
      }

      // serial front-to-back blend over the 16 gaussians of this group
#pragma unroll
      for (int k = 0; k < 16; ++k) {
        const int gg = g0 + k;
        float al = sOp[gg] * __expf(pw[k]);
        al = fminf(al, ALPHA_MAX);
        al = (al < ALPHA_MIN) ? 0.0f : al;
        const float w = al * T;
        aR += sCr[gg] * w;
        aG += sCg[gg] * w;
        aB += sCb[gg] * w;
        T *= (1.0f - al);
      }
    }

    // ---- block-uniform early-out when tile is opaque ----------------------
    float tm = fmaxf(T, xswap16(T));
#pragma unroll
    for (int o2 = 8; o2 > 0; o2 >>= 1) tm = fmaxf(tm, __shfl_xor(tm, o2, 32));
    if (lane == 0) sred[wv] = tm;
    __syncthreads();
    float mT = 0.0f;
#pragma unroll
    for (int w2 = 0; w2 < 8; ++w2) mT = fmaxf(mT, sred[w2]);
    if (mT < 1e-7f) break;
  }

  // ---- composite background, write [3,H,W] --------------------------------
  const int x = tileX * 16 + col;
  const int y = tileY * 16 + row;
  const int pix = y * IMG_W + x;
  const int HW = IMG_W * IMG_H;
  const float b0 = bg[0], b1 = bg[1], b2 = bg[2];
  out[pix] = aR + b0 * T;
  out[HW + pix] = aG + b1 * T;
  out[2 * HW + pix] = aB + b2 * T;
}

// ----------------------------------------------------------------------------
extern "C" void kernel_launch(void* const* d_in, const int* in_sizes, int n_in,
                              void* d_out, int out_size, void* d_ws, size_t ws_size,
                              hipStream_t stream) {
  const float* means  = (const float*)d_in[0];   // [N,2]
  const float* colors = (const float*)d_in[1];   // [N,3]
  const float* opac   = (const float*)d_in[2];   // [N,1]
  const float* scales = (const float*)d_in[3];   // [N,2]
  const float* rots   = (const float*)d_in[4];   // [N,1]
  const float* depths = (const float*)d_in[5];   // [N]
  const float* bg     = (const float*)d_in[6];   // [3]
  int n = in_sizes[5];
  if (n > SORTN) n = SORTN;

  unsigned* order = (unsigned*)d_ws;             // 32 KB

  gs_sort<<<1, 1024, 0, stream>>>(depths, n, order);
  gs_raster<<<dim3((IMG_W / 16) * (IMG_H / 16)), dim3(256), 0, stream>>>(
      means, colors, opac, scales, rots, order, bg, (float*)d_out, n);
}